// GNNEncoder_8100308320889
// MI455X (gfx1250) — compile-verified
//
#include <hip/hip_runtime.h>

#define NN 50000
#define NE 800000
#define IND 128
#define HEADS 4
#define CH 16
#define HCN 64
#define NG 64
#define NEG_SLOPE 0.2f

typedef float v2f __attribute__((ext_vector_type(2)));
typedef float v8f __attribute__((ext_vector_type(8)));

// ---------- helpers: monotone uint encoding of float for atomicMax ----------
__device__ __forceinline__ unsigned fenc(float f) {
    unsigned u = __float_as_uint(f);
    return (u & 0x80000000u) ? ~u : (u | 0x80000000u);
}
__device__ __forceinline__ float fdec(unsigned u) {
    return __uint_as_float((u & 0x80000000u) ? (u & 0x7FFFFFFFu) : ~u);
}

// ---------- zero fill ----------
__global__ void k_zero(float* __restrict__ p, int n) {
    int i = blockIdx.x * blockDim.x + threadIdx.x;
    if (i < n) p[i] = 0.0f;
}

// ---------- WMMA f32 GEMM: C[M,64] = A[M,K] @ B[K,64] ----------
// one wave per 16x16 tile, 4 waves (128 thr) per block -> 16 rows x 64 cols per block.
// M must be a multiple of 16 (50000 = 16*3125). EXEC is all-1s everywhere.
__global__ void k_gemm_wmma(const float* __restrict__ A, const float* __restrict__ B,
                            float* __restrict__ C, int K) {
    const int lane  = threadIdx.x & 31;
    const int wave  = threadIdx.x >> 5;          // N-tile 0..3
    const int row0  = blockIdx.x * 16;
    const int col0  = wave * 16;
    const int r     = lane & 15;
    const int khalf = (lane >> 4) * 2;           // 0 or 2

    v8f acc = {};
    const float* ap = A + (size_t)(row0 + r) * K + khalf;
    const float* bp = B + (size_t)khalf * HCN + col0 + r;
    for (int k0 = 0; k0 < K; k0 += 4) {
        v2f a, b;
        a.x = ap[0];          // A[row][k0+khalf]
        a.y = ap[1];          // A[row][k0+khalf+1]
        b.x = bp[0];          // B[k0+khalf][col]
        b.y = bp[HCN];        // B[k0+khalf+1][col]
        ap += 4;
        bp += 4 * HCN;
        acc = __builtin_amdgcn_wmma_f32_16x16x4_f32(
            /*neg_a=*/false, a, /*neg_b=*/false, b,
            /*c_mod=*/(short)0, acc, /*reuse_a=*/false, /*reuse_b=*/false);
    }
    // C layout: VGPR v -> (lane<16 ? row v : row v+8), col = lane&15
    const int rbase = row0 + (lane >> 4) * 8;
    const int c     = col0 + r;
#pragma unroll
    for (int v = 0; v < 8; ++v)
        C[(size_t)(rbase + v) * HCN + c] = acc[v];
}

// ---------- per-node attention scalars: a_src[n,h], a_dst[n,h] ----------
__global__ void k_attn_coef(const float* __restrict__ xs,
                            const float* __restrict__ att_s,   // [H,CH]
                            const float* __restrict__ att_d,   // [H,CH]
                            float* __restrict__ asrc, float* __restrict__ adst) {
    int idx = blockIdx.x * blockDim.x + threadIdx.x;
    if (idx >= NN * HEADS) return;
    int n = idx >> 2, h = idx & 3;
    const float* row = xs + (size_t)n * HCN + h * CH;
    float s = 0.f, d = 0.f;
#pragma unroll
    for (int c = 0; c < CH; ++c) {
        float v = row[c];
        s += v * att_s[h * CH + c];
        d += v * att_d[h * CH + c];
    }
    asrc[idx] = s;
    adst[idx] = d;
}

// ---------- pass 1: alpha = lrelu(a_src[src]+a_dst[dst]+a_edge), segmax ----------
__global__ void k_alpha_max(const int* __restrict__ src, const int* __restrict__ dst,
                            const float* __restrict__ eattr,
                            const float* __restrict__ asrc, const float* __restrict__ adst,
                            const float* __restrict__ We,  // [1,HC]
                            const float* __restrict__ ae,  // [H,CH]
                            float* __restrict__ alphabuf,
                            unsigned* __restrict__ segmax) {
    int idx = blockIdx.x * blockDim.x + threadIdx.x;
    if (idx >= NE * HEADS) return;
    int e = idx >> 2, h = idx & 3;
    float ce = 0.f;
#pragma unroll
    for (int c = 0; c < CH; ++c) ce += We[h * CH + c] * ae[h * CH + c];
    int sN = src[e], dN = dst[e];
    float a = asrc[sN * HEADS + h] + adst[dN * HEADS + h] + eattr[e] * ce;
    a = a > 0.f ? a : NEG_SLOPE * a;
    alphabuf[idx] = a;
    atomicMax(&segmax[dN * HEADS + h], fenc(a));
}

// ---------- pass 2: ealpha = exp(alpha - segmax[dst]), segsum ----------
__global__ void k_exp_sum(const int* __restrict__ dst,
                          const unsigned* __restrict__ segmax,
                          float* __restrict__ alphabuf,
                          float* __restrict__ segsum) {
    int idx = blockIdx.x * blockDim.x + threadIdx.x;
    if (idx >= NE * HEADS) return;
    int e = idx >> 2, h = idx & 3;
    int dN = dst[e];
    float ea = __expf(alphabuf[idx] - fdec(segmax[dN * HEADS + h]));
    alphabuf[idx] = ea;
    atomicAdd(&segsum[dN * HEADS + h], ea);
}

// ---------- pass 3: out[dst] += coef * xs[src] ----------
__global__ void k_scatter(const int* __restrict__ src, const int* __restrict__ dst,
                          const float* __restrict__ alphabuf,
                          const float* __restrict__ segsum,
                          const float* __restrict__ xs,
                          float* __restrict__ outh) {
    int idx = blockIdx.x * blockDim.x + threadIdx.x;
    if (idx >= NE * HEADS) return;
    int e = idx >> 2, h = idx & 3;
    int sN = src[e], dN = dst[e];
    float coef = alphabuf[idx] / segsum[dN * HEADS + h];
    const float4* xr = (const float4*)(xs + (size_t)sN * HCN + h * CH);
    float* o = outh + (size_t)dN * HCN + h * CH;
#pragma unroll
    for (int i = 0; i < 4; ++i) {
        float4 v = xr[i];
        atomicAdd(o + 4 * i + 0, v.x * coef);
        atomicAdd(o + 4 * i + 1, v.y * coef);
        atomicAdd(o + 4 * i + 2, v.z * coef);
        atomicAdd(o + 4 * i + 3, v.w * coef);
    }
}

// ---------- bias (+ optional relu) ----------
__global__ void k_bias_act(const float* __restrict__ in, const float* __restrict__ bias,
                           float* __restrict__ out, int do_relu) {
    int idx = blockIdx.x * blockDim.x + threadIdx.x;
    if (idx >= NN * HCN) return;
    float v = in[idx] + bias[idx & (HCN - 1)];
    out[idx] = do_relu ? fmaxf(v, 0.f) : v;
}

// ---------- global mean pool: accumulate ----------
__global__ void k_pool_acc(const float* __restrict__ hbuf, const float* __restrict__ b2,
                           const int* __restrict__ batch,
                           float* __restrict__ pool, float* __restrict__ cnt) {
    int idx = blockIdx.x * blockDim.x + threadIdx.x;
    if (idx >= NN * HCN) return;
    int n = idx >> 6, j = idx & 63;
    int g = batch[n];
    atomicAdd(&pool[g * HCN + j], hbuf[idx] + b2[j]);
    if (j == 0) atomicAdd(&cnt[g], 1.0f);
}

__global__ void k_pool_div(const float* __restrict__ pool, const float* __restrict__ cnt,
                           float* __restrict__ out) {
    int idx = blockIdx.x * blockDim.x + threadIdx.x;
    if (idx >= NG * HCN) return;
    out[idx] = pool[idx] / fmaxf(cnt[idx >> 6], 1.0f);
}

extern "C" void kernel_launch(void* const* d_in, const int* in_sizes, int n_in,
                              void* d_out, int out_size, void* d_ws, size_t ws_size,
                              hipStream_t stream) {
    const float* x     = (const float*)d_in[0];
    const int*   ei    = (const int*)d_in[1];
    const float* eattr = (const float*)d_in[2];
    const int*   batch = (const int*)d_in[3];
    const float* W1  = (const float*)d_in[4];
    const float* We1 = (const float*)d_in[5];
    const float* as1 = (const float*)d_in[6];
    const float* ad1 = (const float*)d_in[7];
    const float* ae1 = (const float*)d_in[8];
    const float* b1  = (const float*)d_in[9];
    const float* W2  = (const float*)d_in[10];
    const float* We2 = (const float*)d_in[11];
    const float* as2 = (const float*)d_in[12];
    const float* ad2 = (const float*)d_in[13];
    const float* ae2 = (const float*)d_in[14];
    const float* b2  = (const float*)d_in[15];
    float* out = (float*)d_out;

    const int* src = ei;        // edge_index[0]
    const int* dst = ei + NE;   // edge_index[1]

    // ---- workspace layout (floats) ----
    float*    ws      = (float*)d_ws;
    float*    buf_h   = ws;                                  // N*HC  (aggregation target)
    float*    segsum  = buf_h + (size_t)NN * HCN;            // N*H
    unsigned* segmax  = (unsigned*)(segsum + NN * HEADS);    // N*H
    float*    pool    = (float*)(segmax + NN * HEADS);       // G*HC
    float*    cnt     = pool + NG * HCN;                     // G
    float*    buf_xs  = cnt + NG;                            // N*HC  (xs of current layer)
    float*    buf_hin = buf_xs + (size_t)NN * HCN;           // N*HC  (relu'd layer-1 out)
    float*    asrc    = buf_hin + (size_t)NN * HCN;          // N*H
    float*    adst    = asrc + NN * HEADS;                   // N*H
    float*    alphab  = adst + NN * HEADS;                   // E*H

    const int ZN = NN * HCN + 2 * NN * HEADS + NG * HCN + NG; // buf_h..cnt contiguous
    const int TPB = 256;
    const int gZ  = (ZN + TPB - 1) / TPB;
    const int gNH = (NN * HEADS + TPB - 1) / TPB;
    const int gEH = (NE * HEADS + TPB - 1) / TPB;
    const int gNC = (NN * HCN + TPB - 1) / TPB;
    const int gMT = NN / 16;   // 3125 row tiles

    // ================= layer 1 =================
    k_zero<<<gZ, TPB, 0, stream>>>(buf_h, ZN);   // buf_h, segsum, segmax(=0 == -inf enc), pool, cnt
    k_gemm_wmma<<<gMT, 128, 0, stream>>>(x, W1, buf_xs, IND);
    k_attn_coef<<<gNH, TPB, 0, stream>>>(buf_xs, as1, ad1, asrc, adst);
    k_alpha_max<<<gEH, TPB, 0, stream>>>(src, dst, eattr, asrc, adst, We1, ae1, alphab, segmax);
    k_exp_sum<<<gEH, TPB, 0, stream>>>(dst, segmax, alphab, segsum);
    k_scatter<<<gEH, TPB, 0, stream>>>(src, dst, alphab, segsum, buf_xs, buf_h);
    k_bias_act<<<gNC, TPB, 0, stream>>>(buf_h, b1, buf_hin, /*relu=*/1);

    // ================= layer 2 =================
    k_zero<<<gZ, TPB, 0, stream>>>(buf_h, ZN);
    k_gemm_wmma<<<gMT, 128, 0, stream>>>(buf_hin, W2, buf_xs, HCN);
    k_attn_coef<<<gNH, TPB, 0, stream>>>(buf_xs, as2, ad2, asrc, adst);
    k_alpha_max<<<gEH, TPB, 0, stream>>>(src, dst, eattr, asrc, adst, We2, ae2, alphab, segmax);
    k_exp_sum<<<gEH, TPB, 0, stream>>>(dst, segmax, alphab, segsum);
    k_scatter<<<gEH, TPB, 0, stream>>>(src, dst, alphab, segsum, buf_xs, buf_h);

    // ================= global mean pool =================
    k_pool_acc<<<gNC, TPB, 0, stream>>>(buf_h, b2, batch, pool, cnt);
    k_pool_div<<<(NG * HCN + TPB - 1) / TPB, TPB, 0, stream>>>(pool, cnt, out);
}